// WrappedSequencedUnstridedTemporalConv1d_50869592655589
// MI455X (gfx1250) — compile-verified
//
#include <hip/hip_runtime.h>
#include <hip/hip_bf16.h>

// Causal dilated conv1d (N=8, Cin=512, L=4096, Cout=512, K=4, dil=2) as an
// implicit GEMM on fp32 WMMA (V_WMMA_F32_16X16X4_F32).
//
// HBM-bound (~132 MB moved vs ~69 GFLOP -> ~5.7us floor at 23.3 TB/s):
// stream X/Y exactly once, let the 192MB L2 hold the 4MB weight, use the
// matrix pipe so math never limits. Each wave computes a 32(Cout) x 32(t)
// tile as a 2x2 outer product of fragments: 6 VMEM instructions feed 4 WMMAs
// (1.5 VMEM/WMMA) and the 4 accumulators form independent dependency chains,
// so consecutive v_wmma issues never hit the D->C RAW hazard.

typedef __attribute__((ext_vector_type(2))) float v2f;
typedef __attribute__((ext_vector_type(8))) float v8f;

namespace {
constexpr int kN    = 8;
constexpr int kCin  = 512;
constexpr int kL    = 4096;
constexpr int kCout = 512;
constexpr int kTaps = 4;
constexpr int kDil  = 2;
constexpr int kKdim = kTaps * kCin;            // 2048
constexpr int kPad  = (kTaps - 1) * kDil;      // 6 (causal left reach)
}

__global__ __launch_bounds__(256)
void conv1d_wmma_f32(const float* __restrict__ X,   // (N, Cin, L)
                     const float* __restrict__ W,   // (Cout, K*Cin) taps oldest->newest
                     const float* __restrict__ Bias,// (Cout)
                     float* __restrict__ Y)         // (N, Cout, L)
{
    const int lane  = threadIdx.x & 31;
    const int wave  = threadIdx.x >> 5;
    const int col   = lane & 15;        // t column within 16-wide sub-tile / A-row m
    const int khalf = lane >> 4;        // which K-half of the 16x4 fragment

    // 8 waves: 2 Cout strips (32) x 4 t strips (32) -> block tile 64 x 128
    const int co0 = blockIdx.y * 64  + (wave & 1) * 32;
    const int t0  = blockIdx.x * 128 + (wave >> 1) * 32;
    const int n   = blockIdx.z;

    const float* xn = X + (size_t)n * kCin * kL;
    // A layout (32-bit 16x4): lane l -> row m=l&15; VGPR0/1 hold K = 2*(l>>4)+{0,1}
    const float* wrow0 = W + (size_t)(co0 + col) * kKdim + 2 * khalf;
    const float* wrow1 = wrow0 + (size_t)16 * kKdim;   // second Cout strip of 16

    v8f acc00 = {};   // (co0..+15 , t0..+15)
    v8f acc01 = {};   // (co0..+15 , t0+16..+31)
    v8f acc10 = {};   // (co0+16..+31, t0..+15)
    v8f acc11 = {};   // (co0+16..+31, t0+16..+31)

    if (t0 >= kPad) {
        // ---- interior fast path: loads feed WMMA directly, no VALU ----
        #pragma unroll
        for (int j = 0; j < kTaps; ++j) {
            const float* xc0 = xn + (t0 + col - (kTaps - 1 - j) * kDil);
            const float* xc1 = xc0 + 16;
            const float* wj0 = wrow0 + j * kCin;
            const float* wj1 = wrow1 + j * kCin;

            #pragma unroll 4
            for (int ci = 0; ci < kCin; ci += 4) {
                v2f a0 = *(const v2f*)(wj0 + ci);      // global_load_b64
                v2f a1 = *(const v2f*)(wj1 + ci);      // global_load_b64
                // B layout (4x16): VGPR0 K={0,1} on lane halves, VGPR1 K={2,3}
                v2f b0, b1;
                b0.x = xc0[(size_t)(ci + khalf)     * kL];
                b0.y = xc0[(size_t)(ci + 2 + khalf) * kL];
                b1.x = xc1[(size_t)(ci + khalf)     * kL];
                b1.y = xc1[(size_t)(ci + 2 + khalf) * kL];
                // 2x2 outer product: 4 independent accumulator chains
                acc00 = __builtin_amdgcn_wmma_f32_16x16x4_f32(
                    false, a0, false, b0, (short)0, acc00, false, false);
                acc10 = __builtin_amdgcn_wmma_f32_16x16x4_f32(
                    false, a1, false, b0, (short)0, acc10, false, false);
                acc01 = __builtin_amdgcn_wmma_f32_16x16x4_f32(
                    false, a0, false, b1, (short)0, acc01, false, false);
                acc11 = __builtin_amdgcn_wmma_f32_16x16x4_f32(
                    false, a1, false, b1, (short)0, acc11, false, false);
            }
        }
    } else {
        // ---- left edge (t0 == 0 strip only): branch-free per-lane masking ----
        #pragma unroll
        for (int j = 0; j < kTaps; ++j) {
            const int shift = (kTaps - 1 - j) * kDil;
            const int tA0 = t0 + col - shift;          // can underflow
            const int tA1 = t0 + 16 + col - shift;     // never underflows (>=10)
            const float m0 = (tA0 >= 0) ? 1.0f : 0.0f;
            const float* xc0 = xn + ((tA0 >= 0) ? tA0 : 0);
            const float* xc1 = xn + tA1;
            const float* wj0 = wrow0 + j * kCin;
            const float* wj1 = wrow1 + j * kCin;

            #pragma unroll 2
            for (int ci = 0; ci < kCin; ci += 4) {
                v2f a0 = *(const v2f*)(wj0 + ci);
                v2f a1 = *(const v2f*)(wj1 + ci);
                v2f b0, b1;
                b0.x = m0 * xc0[(size_t)(ci + khalf)     * kL];
                b0.y = m0 * xc0[(size_t)(ci + 2 + khalf) * kL];
                b1.x = xc1[(size_t)(ci + khalf)     * kL];
                b1.y = xc1[(size_t)(ci + 2 + khalf) * kL];
                acc00 = __builtin_amdgcn_wmma_f32_16x16x4_f32(
                    false, a0, false, b0, (short)0, acc00, false, false);
                acc10 = __builtin_amdgcn_wmma_f32_16x16x4_f32(
                    false, a1, false, b0, (short)0, acc10, false, false);
                acc01 = __builtin_amdgcn_wmma_f32_16x16x4_f32(
                    false, a0, false, b1, (short)0, acc01, false, false);
                acc11 = __builtin_amdgcn_wmma_f32_16x16x4_f32(
                    false, a1, false, b1, (short)0, acc11, false, false);
            }
        }
    }

    // D layout: VGPR r -> row M = r + 8*(lane>>4), col N = lane&15
    float* yn = Y + (size_t)n * kCout * kL;
    #pragma unroll
    for (int r = 0; r < 8; ++r) {
        const int co = co0 + r + 8 * khalf;
        const float ba = Bias[co];
        const float bb = Bias[co + 16];
        yn[(size_t)co * kL        + t0 + col]      = acc00[r] + ba;
        yn[(size_t)co * kL        + t0 + 16 + col] = acc01[r] + ba;
        yn[(size_t)(co + 16) * kL + t0 + col]      = acc10[r] + bb;
        yn[(size_t)(co + 16) * kL + t0 + 16 + col] = acc11[r] + bb;
    }
}

extern "C" void kernel_launch(void* const* d_in, const int* in_sizes, int n_in,
                              void* d_out, int out_size, void* d_ws, size_t ws_size,
                              hipStream_t stream) {
    const float* x    = (const float*)d_in[0];
    const float* w    = (const float*)d_in[1];
    const float* bias = (const float*)d_in[2];
    // d_in[3] is dilation (==2), baked in as a compile-time constant.
    float* y = (float*)d_out;

    dim3 grid(kL / 128, kCout / 64, kN);   // 32 x 8 x 8 = 2048 blocks
    conv1d_wmma_f32<<<grid, 256, 0, stream>>>(x, w, bias, y);
}